// LayerStacks_8169027797240
// MI455X (gfx1250) — compile-verified
//
#include <hip/hip_runtime.h>

typedef float v2f __attribute__((ext_vector_type(2)));
typedef float v8f __attribute__((ext_vector_type(8)));
typedef int   v4i_vs __attribute__((vector_size(16)));   // gcc-style int4 (builtin param type)
typedef __attribute__((address_space(1))) v4i_vs* gptr_v4i;
typedef __attribute__((address_space(3))) v4i_vs* lptr_v4i;

#define L1D   3072
#define COUNT 8
#define NW    144   // 128 (l1_w rows) + 16 (l1f_w rows) fused output columns
#define BK    32    // K chunk staged in LDS
#define XS    36    // padded LDS stride (conflict-free frag reads)
#define NCH   (L1D / BK)   // 96 K-chunks

#if __has_builtin(__builtin_amdgcn_global_load_async_to_lds_b128) && \
    __has_builtin(__builtin_amdgcn_s_wait_asynccnt)
#define USE_ASYNC_LDS 1
#else
#define USE_ASYNC_LDS 0
#endif

__device__ __forceinline__ void cp16_g2lds(const float* g, float* l)
{
#if USE_ASYNC_LDS
    __builtin_amdgcn_global_load_async_to_lds_b128(
        (gptr_v4i)(g), (lptr_v4i)(l), 0, 0);
#else
    *reinterpret_cast<float4*>(l) = *reinterpret_cast<const float4*>(g);
#endif
}

struct GemmBufs { float xb[2][16 * XS]; float wb[2][NW * XS]; }; // 46.1 KB
struct KanBufs  { float silu[16 * 30]; float basis[16 * 30 * 6]; }; // 13.4 KB

__global__ __launch_bounds__(288)
void fused_layerstacks_kernel(const float* __restrict__ x,
                              const int*   __restrict__ ls_indices,
                              const float* __restrict__ l1_w,
                              const float* __restrict__ l1_b,
                              const float* __restrict__ l1f_w,
                              const float* __restrict__ l1f_b,
                              const float* __restrict__ kan_grid,
                              const float* __restrict__ kan_coef,
                              const float* __restrict__ kan_scale_base,
                              const float* __restrict__ kan_scale_sp,
                              const float* __restrict__ out_w,
                              const float* __restrict__ out_b,
                              float* __restrict__ out)
{
    __shared__ union { GemmBufs g; KanBufs k; } sm;   // time-multiplexed
    __shared__ float s_res[16 * NW];                  // GEMM results + bias
    __shared__ float s_bias[NW];

    const int t    = threadIdx.x;
    const int wave = t >> 5;          // 0..8 -> N-tile id
    const int lane = t & 31;
    const int m16  = lane & 15;
    const int koff = (lane < 16) ? 0 : 2;   // fp32 WMMA frag K-offset per half-wave
    const int rowBase = blockIdx.x * 16;

    if (t < NW) s_bias[t] = (t < 128) ? l1_b[t] : l1f_b[t - 128];

    // per-thread staging coordinates
    const int xr = t >> 3, xc = (t & 7) * 4;            // x tile (threads 0..127)
    const float* xsrc = x + (size_t)(rowBase + xr) * L1D + xc;

    v8f c = {};   // 16x16 f32 accumulator (8 VGPRs)

    // issue one K-chunk's copies into LDS buffer `buf`
    auto issue_chunk = [&](int kb, int buf) {
        if (wave < 4)                                   // t < 128
            cp16_g2lds(xsrc + kb, &sm.g.xb[buf][xr * XS + xc]);
        #pragma unroll
        for (int jj = 0; jj < 4; ++jj) {
            const int j = t + 288 * jj;                 // 1152 float4 total
            const int r = j >> 3, cc = (j & 7) * 4;
            const float* wp = (r < 128) ? (l1_w + (size_t)r * L1D)
                                        : (l1f_w + (size_t)(r - 128) * L1D);
            cp16_g2lds(wp + kb + cc, &sm.g.wb[buf][r * XS + cc]);
        }
    };

    // ------------- Stage 1: double-buffered fused GEMM [16x3072]@[3072x144] --
    issue_chunk(0, 0);
    for (int ch = 0; ch < NCH; ++ch) {
        const int  cur  = ch & 1;
        const bool more = (ch + 1 < NCH);
        if (more) issue_chunk((ch + 1) * BK, cur ^ 1);
#if USE_ASYNC_LDS
        if (more) {                                     // wait: only next chunk in flight
            if (wave < 4) __builtin_amdgcn_s_wait_asynccnt(5);
            else          __builtin_amdgcn_s_wait_asynccnt(4);
        } else {
            __builtin_amdgcn_s_wait_asynccnt(0);
        }
#endif
        __syncthreads();                                // chunk `ch` visible to all

        if (more && wave < 4)                           // warm L2 two chunks ahead
            __builtin_prefetch(xsrc + (ch + 2) * BK, 0, 3);

        const float* xrow = &sm.g.xb[cur][m16 * XS];
        const float* wrow = &sm.g.wb[cur][(wave * 16 + m16) * XS];
        #pragma unroll
        for (int kk = 0; kk < BK; kk += 4) {
            v2f a = *reinterpret_cast<const v2f*>(xrow + kk + koff);
            v2f b = *reinterpret_cast<const v2f*>(wrow + kk + koff);
            c = __builtin_amdgcn_wmma_f32_16x16x4_f32(
                    false, a, false, b, (short)0, c, false, false);
        }
        __syncthreads();                                // done reading buf `cur`
    }

    // write D (+bias) to LDS: VGPR r -> M = r (lanes 0-15) / r+8 (lanes 16-31)
    {
        const int col = wave * 16 + m16;
        const float bv = s_bias[col];
        const int r0 = (lane < 16) ? 0 : 8;
        #pragma unroll
        for (int r = 0; r < 8; ++r)
            s_res[(r0 + r) * NW + col] = c[r] + bv;
    }
    __syncthreads();

    // ------------- Stage 2: gather + KAN + output head -----------------------
    for (int pass = 0; pass < 2; ++pass) {
        const int rr  = wave + 9 * pass;     // row within M-tile handled by wave
        const bool act = (rr < 16);
        const int grow = rowBase + rr;
        int sel = 0;
        if (act) sel = ls_indices[grow];

        // Phase A: lanes 0..29 build features, SiLU, cubic B-spline basis
        if (act && lane < 30) {
            const int i  = lane;
            const int im = (i < 15) ? i : (i - 15);
            const float tv = s_res[rr * NW + sel * 16 + im]
                           + s_res[rr * NW + 128 + im];
            float f = (i < 15) ? tv * tv * (127.0f / 128.0f) : tv;
            f = fminf(fmaxf(f, 0.0f), 1.0f);
            sm.k.silu[rr * 30 + i] = f / (1.0f + __expf(-f));

            float g[10];
            #pragma unroll
            for (int m = 0; m < 10; ++m) g[m] = kan_grid[i * 10 + m];
            float Bv[9];
            #pragma unroll
            for (int m = 0; m < 9; ++m)
                Bv[m] = (f >= g[m] && f < g[m + 1]) ? 1.0f : 0.0f;
            #pragma unroll
            for (int d = 1; d <= 3; ++d) {
                #pragma unroll
                for (int m = 0; m < 9 - d; ++m) {
                    Bv[m] = (f - g[m]) / (g[m + d] - g[m]) * Bv[m]
                          + (g[m + d + 1] - f) / (g[m + d + 1] - g[m + 1]) * Bv[m + 1];
                }
            }
            #pragma unroll
            for (int k = 0; k < 6; ++k)
                sm.k.basis[(rr * 30 + i) * 6 + k] = Bv[k];
        }
        __syncthreads();

        // Phase B: each lane computes one of the 32 selected KAN outputs
        if (act) {
            const int o = sel * 32 + lane;
            float acc = 0.0f;
            for (int i = 0; i < 30; ++i) {
                const float sl = sm.k.silu[rr * 30 + i];
                const float sb = kan_scale_base[i * 256 + o];
                const float sp = kan_scale_sp[i * 256 + o];
                const float* cp = kan_coef + (size_t)(i * 256 + o) * 6;
                float bs = 0.0f;
                #pragma unroll
                for (int k = 0; k < 6; ++k)
                    bs += sm.k.basis[(rr * 30 + i) * 6 + k] * cp[k];
                acc += sl * sb + bs * sp;
            }
            acc = fminf(fmaxf(acc, 0.0f), 1.0f);
            float p = acc * out_w[o];
            #pragma unroll
            for (int off = 16; off > 0; off >>= 1)
                p += __shfl_xor(p, off, 32);      // wave32 reduction
            if (lane == 0) {
                const float extra = s_res[rr * NW + sel * 16 + 15]
                                  + s_res[rr * NW + 128 + 15];
                out[grow] = p + out_b[sel] + extra;
            }
        }
        __syncthreads();
    }
}

extern "C" void kernel_launch(void* const* d_in, const int* in_sizes, int n_in,
                              void* d_out, int out_size, void* d_ws, size_t ws_size,
                              hipStream_t stream)
{
    (void)n_in; (void)d_ws; (void)ws_size; (void)out_size;
    const float* x       = (const float*)d_in[0];
    const int*   ls_idx  = (const int*)  d_in[1];
    const float* l1_w    = (const float*)d_in[2];
    const float* l1_b    = (const float*)d_in[3];
    const float* l1f_w   = (const float*)d_in[4];
    const float* l1f_b   = (const float*)d_in[5];
    const float* kgrid   = (const float*)d_in[6];
    const float* kcoef   = (const float*)d_in[7];
    const float* ksb     = (const float*)d_in[8];
    const float* ksp     = (const float*)d_in[9];
    const float* out_w   = (const float*)d_in[10];
    const float* out_b   = (const float*)d_in[11];
    float*       out     = (float*)d_out;

    const int B = in_sizes[1];              // 16384
    dim3 grid(B / 16), block(288);
    fused_layerstacks_kernel<<<grid, block, 0, stream>>>(
        x, ls_idx, l1_w, l1_b, l1f_w, l1f_b,
        kgrid, kcoef, ksb, ksp, out_w, out_b, out);
}